// FineMatching_66477503808168
// MI455X (gfx1250) — compile-verified
//
#include <hip/hip_runtime.h>
#include <cstdint>

#define WW 25          // W*W window
#define CH 128         // channels
#define ROW_PAD 132    // 128 dwords + 4 pad -> conflict-free ds_load_b64 pattern
#define WAVES_PER_BLOCK 4
#define WAVE_LDS (26 * ROW_PAD)   // 25 K rows + 1 query row (dwords)

typedef float v2f __attribute__((ext_vector_type(2)));
typedef float v8f __attribute__((ext_vector_type(8)));

// CDNA5 async DMA: global memory -> LDS, 16B per lane, tracked by ASYNCcnt.
__device__ __forceinline__ void async_b128(uint32_t lds_byte, const void* gaddr) {
  asm volatile("global_load_async_to_lds_b128 %0, %1, off"
               :: "v"(lds_byte), "v"(gaddr)
               : "memory");
}

__device__ __forceinline__ void wait_async0() {
#if __has_builtin(__builtin_amdgcn_s_wait_asynccnt)
  __builtin_amdgcn_s_wait_asynccnt(0);
#else
  asm volatile("s_wait_asynccnt 0" ::: "memory");
#endif
}

__global__ __launch_bounds__(WAVES_PER_BLOCK * 32)
void FineMatching_kernel(const float* __restrict__ feat_db,   // (M,3,128)
                         const float* __restrict__ feat_q,    // (M,25,128)
                         const float* __restrict__ mkpts,     // (M,2)
                         float* __restrict__ out,             // [3M expec | 2M mkpts_f]
                         int M) {
  __shared__ float smem[WAVES_PER_BLOCK * WAVE_LDS];

  const int lane = threadIdx.x & 31;
  const int wave = threadIdx.x >> 5;
  const int m    = blockIdx.x * WAVES_PER_BLOCK + wave;
  if (m >= M) return;   // grid sized exactly: every live wave has full EXEC

  float* wbase = &smem[wave * WAVE_LDS];
  const uint32_t wbyte = (uint32_t)(uintptr_t)wbase;  // low 32 bits = LDS offset

  // ---- Stage K tile (25x128 f32) + picked query row into LDS (async DMA) ----
  const float* gK = feat_q  + (size_t)m * (WW * CH);
  const float* gq = feat_db + (size_t)m * (3 * CH) + CH;   // feat_db_3d[m, 1, :]
  #pragma unroll
  for (int r = 0; r < WW; ++r) {
    async_b128(wbyte + (uint32_t)(r * ROW_PAD + lane * 4) * 4u,
               gK + r * CH + lane * 4);
  }
  async_b128(wbyte + (uint32_t)(WW * ROW_PAD + lane * 4) * 4u, gq + lane * 4);
  wait_async0();

  // ---- sim = K(25x128) @ q(128) via V_WMMA_F32_16X16X4_F32 ----
  // A 16x4 f32 layout: lanes 0-15 hold K=0,1 (VGPR0,1); lanes 16-31 hold K=2,3.
  // B 4x16: same K split across lane halves; we broadcast q to all 16 columns.
  const int half = lane >> 4;
  const int rlo  = lane & 15;
  const int rowA1 = (rlo > 8) ? 24 : (16 + rlo);  // tile1 rows 16..24, dup-clamped

  const float* A0p = wbase + rlo   * ROW_PAD + 2 * half;
  const float* A1p = wbase + rowA1 * ROW_PAD + 2 * half;
  const float* Bp  = wbase + WW    * ROW_PAD + 2 * half;

  v8f acc0 = {0.f,0.f,0.f,0.f,0.f,0.f,0.f,0.f};
  v8f acc1 = {0.f,0.f,0.f,0.f,0.f,0.f,0.f,0.f};
  #pragma unroll
  for (int i = 0; i < CH / 4; ++i) {
    v2f a0 = *(const v2f*)(A0p + 4 * i);
    v2f a1 = *(const v2f*)(A1p + 4 * i);
    v2f b  = *(const v2f*)(Bp  + 4 * i);
    acc0 = __builtin_amdgcn_wmma_f32_16x16x4_f32(false, a0, false, b,
                                                 (short)0, acc0, false, false);
    acc1 = __builtin_amdgcn_wmma_f32_16x16x4_f32(false, a1, false, b,
                                                 (short)0, acc1, false, false);
  }

  // ---- Gather all 25 sims into every lane (D columns are identical) ----
  // D layout: VGPR v -> M=v (lanes 0-15) / M=v+8 (lanes 16-31).
  float s[WW];
  #pragma unroll
  for (int v = 0; v < 8; ++v) {
    float other = __shfl_xor(acc0[v], 16);
    s[half ? (v + 8) : v] = acc0[v];
    s[half ? v : (v + 8)] = other;
  }
  float s24 = 0.f;
  #pragma unroll
  for (int v = 0; v < 8; ++v) {
    float other = __shfl_xor(acc1[v], 16);
    s[16 + v] = half ? other : acc1[v];          // rows 16..23
    if (v == 0) s24 = half ? acc1[0] : other;    // row 24 lives in lanes>=16, VGPR0
  }
  s[24] = s24;

  // ---- softmax(sim / sqrt(C)) and grid moments (grid constants folded) ----
  const float kInvSqrtC = 0.08838834764831845f;  // 1/sqrt(128)
  float mx = s[0];
  #pragma unroll
  for (int r = 1; r < WW; ++r) mx = fmaxf(mx, s[r]);

  float se = 0.f, sx = 0.f, sy = 0.f, sxx = 0.f, syy = 0.f;
  #pragma unroll
  for (int r = 0; r < WW; ++r) {
    const float gx = (float)(r % 5) * 0.5f - 1.0f;
    const float gy = (float)(r / 5) * 0.5f - 1.0f;
    float e = __expf((s[r] - mx) * kInvSqrtC);
    se += e;
    sx  += e * gx;   sy  += e * gy;
    sxx += e * gx * gx; syy += e * gy * gy;
  }
  const float inv = 1.0f / se;
  const float cx = sx * inv, cy = sy * inv;
  const float vx = sxx * inv - cx * cx;
  const float vy = syy * inv - cy * cy;
  const float stdv = sqrtf(fmaxf(vx, 1e-10f)) + sqrtf(fmaxf(vy, 1e-10f));

  if (lane == 0) {
    out[3 * (size_t)m + 0] = cx;
    out[3 * (size_t)m + 1] = cy;
    out[3 * (size_t)m + 2] = stdv;
    const float mk0 = mkpts[2 * (size_t)m + 0];
    const float mk1 = mkpts[2 * (size_t)m + 1];
    float* out2 = out + 3 * (size_t)M;
    out2[2 * (size_t)m + 0] = mk0 + cx * 16.0f;   // (W//2)*SCALE = 16
    out2[2 * (size_t)m + 1] = mk1 + cy * 16.0f;
  }
}

extern "C" void kernel_launch(void* const* d_in, const int* in_sizes, int n_in,
                              void* d_out, int out_size, void* d_ws, size_t ws_size,
                              hipStream_t stream) {
  (void)n_in; (void)out_size; (void)d_ws; (void)ws_size;
  const float* feat_db = (const float*)d_in[0];
  const float* feat_q  = (const float*)d_in[1];
  const float* mkpts   = (const float*)d_in[2];
  float* out = (float*)d_out;
  const int M = in_sizes[2] / 2;          // mkpts_query_c is (M,2)
  const int blocks = (M + WAVES_PER_BLOCK - 1) / WAVES_PER_BLOCK;
  hipLaunchKernelGGL(FineMatching_kernel, dim3(blocks),
                     dim3(WAVES_PER_BLOCK * 32), 0, stream,
                     feat_db, feat_q, mkpts, out, M);
}